// GraphSageLayer_30657476558871
// MI455X (gfx1250) — compile-verified
//
#include <hip/hip_runtime.h>

#define N_NODES 50000
#define D_FEAT  128
#define N_EDGES 600000
#define HIDDEN  256
#define KDIM    256   // 2*D_FEAT

typedef __attribute__((ext_vector_type(2))) float v2f;
typedef __attribute__((ext_vector_type(8))) float v8f;

// ---------------------------------------------------------------------------
// Kernel 1: zero the scatter workspace (sums[N*128] ++ counts[N])
// ---------------------------------------------------------------------------
__global__ void sage_zero(float* __restrict__ ws, int n) {
    int i = blockIdx.x * blockDim.x + threadIdx.x;
    if (i < n) ws[i] = 0.0f;
}

// ---------------------------------------------------------------------------
// Kernel 2: transpose W [KDIM x HIDDEN] -> Wt [HIDDEN x KDIM] so that the two
// K-adjacent B-fragment values per lane are contiguous (b64 loads in GEMM).
// Only 256 KB, runs once per launch.
// ---------------------------------------------------------------------------
__global__ void sage_transpose_w(const float* __restrict__ W,
                                 float* __restrict__ Wt) {
    int i = blockIdx.x * blockDim.x + threadIdx.x;   // i = k*HIDDEN + c
    if (i < KDIM * HIDDEN) {
        int k = i >> 8;          // HIDDEN == 256
        int c = i & 255;
        Wt[(size_t)c * KDIM + k] = W[i];
    }
}

// ---------------------------------------------------------------------------
// Kernel 3: edge scatter. One wave32 per edge: lane l moves float4 at
// feature offset l*4 (128 floats total). L2-resident gather + f32 atomics.
// ---------------------------------------------------------------------------
__global__ void sage_scatter(const float* __restrict__ nodes,
                             const int*   __restrict__ senders,
                             const int*   __restrict__ receivers,
                             float* __restrict__ sums,
                             float* __restrict__ counts) {
    int wave = (blockIdx.x * blockDim.x + threadIdx.x) >> 5;
    int lane = threadIdx.x & 31;
    if (wave >= N_EDGES) return;
    int s = senders[wave];
    int r = receivers[wave];
    const float4 v = *(const float4*)(nodes + (size_t)s * D_FEAT + lane * 4);
    float* dst = sums + (size_t)r * D_FEAT + lane * 4;
    unsafeAtomicAdd(dst + 0, v.x);   // -> global_atomic_add_f32
    unsafeAtomicAdd(dst + 1, v.y);
    unsafeAtomicAdd(dst + 2, v.z);
    unsafeAtomicAdd(dst + 3, v.w);
    if (lane == 0) unsafeAtomicAdd(counts + r, 1.0f);
}

// ---------------------------------------------------------------------------
// Kernel 4: fused mean + concat + GEMM + bias + relu via V_WMMA_F32_16X16X4.
// Each wave: 16 rows x 128 cols of out (8 accumulators), K=256 in steps of 4.
// A 16x4 f32 layout: lanes 0-15 = rows, lane-half selects K pair {0,1}/{2,3},
// VGPR selects within pair. B from transposed W: one b64 per fragment.
// C/D layout: VGPR v -> row v (lanes 0-15) / row v+8 (lanes 16-31).
// ---------------------------------------------------------------------------
__global__ void __launch_bounds__(256)
sage_gemm(const float* __restrict__ sums, const float* __restrict__ counts,
          const float* __restrict__ nodes, const float* __restrict__ Wt,
          const float* __restrict__ bias, float* __restrict__ out) {
    const int lane  = threadIdx.x & 31;
    const int wid   = blockIdx.x * (blockDim.x >> 5) + (threadIdx.x >> 5);
    const int mtile = wid >> 1;            // 0..3124 (50000/16 = 3125 exact)
    if (mtile >= (N_NODES / 16)) return;   // wave-uniform: EXEC stays all-ones
    const int n0    = (wid & 1) * 128;     // column base: 8 tiles of 16

    const int row   = lane & 15;
    const int khalf = lane >> 4;           // 0: K{0,1}, 1: K{2,3}
    const int grow  = mtile * 16 + row;    // this lane's A row

    const float  invc = 1.0f / fmaxf(counts[grow], 1.0f);
    const float* srow = sums  + (size_t)grow * D_FEAT;
    const float* nrow = nodes + (size_t)grow * D_FEAT;

    v8f acc[8] = {};

    // --- K = 0..127 : aggregated-mean half of the concat ---
    for (int k0 = 0; k0 < D_FEAT; k0 += 4) {
        const int ka = k0 + khalf * 2;
        v2f a = *(const v2f*)(srow + ka);   // b64 load
        a *= invc;                          // v_pk_mul_f32
#pragma unroll
        for (int t = 0; t < 8; ++t) {
            const int col = n0 + t * 16 + row;
            const v2f bf = *(const v2f*)(Wt + (size_t)col * KDIM + ka); // b64
            acc[t] = __builtin_amdgcn_wmma_f32_16x16x4_f32(
                false, a, false, bf, (short)0, acc[t], false, false);
        }
    }
    // --- K = 128..255 : self-feature half of the concat ---
    for (int k0 = D_FEAT; k0 < KDIM; k0 += 4) {
        const int ka = k0 + khalf * 2;
        const v2f a = *(const v2f*)(nrow + (ka - D_FEAT));   // b64 load
#pragma unroll
        for (int t = 0; t < 8; ++t) {
            const int col = n0 + t * 16 + row;
            const v2f bf = *(const v2f*)(Wt + (size_t)col * KDIM + ka); // b64
            acc[t] = __builtin_amdgcn_wmma_f32_16x16x4_f32(
                false, a, false, bf, (short)0, acc[t], false, false);
        }
    }

    // --- epilogue: bias + relu, C/D VGPR layout stores ---
#pragma unroll
    for (int t = 0; t < 8; ++t) {
        const int   col = n0 + t * 16 + row;
        const float bb  = bias[col];
#pragma unroll
        for (int v = 0; v < 8; ++v) {
            const int orow = mtile * 16 + v + khalf * 8;
            out[(size_t)orow * HIDDEN + col] = fmaxf(acc[t][v] + bb, 0.0f);
        }
    }
}

// ---------------------------------------------------------------------------
extern "C" void kernel_launch(void* const* d_in, const int* in_sizes, int n_in,
                              void* d_out, int out_size, void* d_ws, size_t ws_size,
                              hipStream_t stream) {
    const float* nodes     = (const float*)d_in[0];
    const int*   senders   = (const int*)  d_in[1];
    const int*   receivers = (const int*)  d_in[2];
    const float* Wm        = (const float*)d_in[3];
    const float* bias      = (const float*)d_in[4];
    float*       out       = (float*)d_out;

    float* sums   = (float*)d_ws;                       // N_NODES * D_FEAT
    float* counts = sums + (size_t)N_NODES * D_FEAT;    // N_NODES
    float* Wt     = counts + N_NODES;                   // HIDDEN * KDIM

    const int zn = N_NODES * D_FEAT + N_NODES;
    sage_zero<<<(zn + 255) / 256, 256, 0, stream>>>(sums, zn);

    sage_transpose_w<<<(KDIM * HIDDEN + 255) / 256, 256, 0, stream>>>(Wm, Wt);

    const long long sthreads = (long long)N_EDGES * 32;
    sage_scatter<<<(int)((sthreads + 255) / 256), 256, 0, stream>>>(
        nodes, senders, receivers, sums, counts);

    const int nwaves = (N_NODES / 16) * 2;   // 6250 wave-tiles (16x128 each)
    sage_gemm<<<(nwaves + 7) / 8, 256, 0, stream>>>(
        sums, counts, nodes, Wt, bias, out);
}